// LogisticRegressionRBF_56203942036129
// MI455X (gfx1250) — compile-verified
//
#include <hip/hip_runtime.h>
#include <hip/hip_bf16.h>

typedef __attribute__((ext_vector_type(2))) float v2f;
typedef __attribute__((ext_vector_type(4))) float v4f;
typedef __attribute__((ext_vector_type(8))) float v8f;

#define KROWS 8192
#define NCOLS 8192
#define MFEAT 64
#define NSPLIT 4
#define WAVES_PER_BLOCK 4

// ---- Prep: pack[j] = { ||x_basis_j||^2 , w[j] }  (computed once, L2-resident)
__global__ __launch_bounds__(256, 1)
void rbf_prep_kernel(const float* __restrict__ xb,
                     const float* __restrict__ w,
                     v2f* __restrict__ pack)
{
    int j = blockIdx.x * 256 + threadIdx.x;
    const float* row = xb + (size_t)j * MFEAT;
    float s = 0.f;
#pragma unroll
    for (int k = 0; k < 16; ++k) {
        v4f q = *(const v4f*)(row + 4 * k);
        s += q.x * q.x + q.y * q.y + q.z * q.z + q.w * q.w;
    }
    v2f p; p.x = s; p.y = w[j];
    pack[j] = p;
}

// ---- Main: wave-per-16-row-tile fused RBF-GEMM over an N-chunk.
__global__ __launch_bounds__(WAVES_PER_BLOCK * 32, 1)
void rbf_partial_kernel(const float* __restrict__ x,
                        const float* __restrict__ xb,
                        const v2f* __restrict__ pack,
                        float* __restrict__ partial /* [NSPLIT][KROWS] */)
{
    const int lane   = threadIdx.x & 31;
    const int wave   = threadIdx.x >> 5;
    const int waveId = blockIdx.x * WAVES_PER_BLOCK + wave;
    const int rowTile = waveId / NSPLIT;     // 0..511
    const int split   = waveId % NSPLIT;     // 0..3
    const int rowBase = rowTile * 16;
    const int half = lane >> 4;              // K-group per f32 WMMA A layout
    const int lrow = lane & 15;

    // A fragments for V_WMMA_F32_16X16X4_F32 (ISA 7.12.2):
    // lane (half*16 + m) holds row m, features kBase + half*2 + {0,1} -> contiguous float2
    const float* xrow = x + (size_t)(rowBase + lrow) * MFEAT + half * 2;
    v2f a[16];
    float xsqPart = 0.f;
#pragma unroll
    for (int kk = 0; kk < 16; ++kk) {
        a[kk] = *(const v2f*)(xrow + kk * 4);
        xsqPart += a[kk].x * a[kk].x + a[kk].y * a[kk].y;
    }
    // complete ||x_row||^2: lanes L and L+16 each hold half of row (L&15)
    float xsq = xsqPart + __shfl_xor(xsqPart, 16, 32);
    // per-C-register row norms: C reg r maps to row r + 8*half
    float xnr[8];
#pragma unroll
    for (int r = 0; r < 8; ++r)
        xnr[r] = __shfl(xsq, r + 8 * half, 32);

    float acc[8];
#pragma unroll
    for (int r = 0; r < 8; ++r) acc[r] = 0.f;

    const int jBeg = split * (NCOLS / NSPLIT);
    const int jEnd = jBeg + (NCOLS / NSPLIT);

    for (int j = jBeg; j < jEnd; j += 32) {
        // B fragments: B[k][n] = xb[col_base + n][kBase + k]; same lane layout as A.
        const float* brow0 = xb + (size_t)(j + lrow) * MFEAT + half * 2;
        const float* brow1 = brow0 + 16 * MFEAT;
        __builtin_prefetch(brow0 + 32 * MFEAT, 0, 1);   // next iteration's basis rows

        v2f b0[16], b1[16];
#pragma unroll
        for (int kk = 0; kk < 16; ++kk) {
            b0[kk] = *(const v2f*)(brow0 + kk * 4);
            b1[kk] = *(const v2f*)(brow1 + kk * 4);
        }
        const v2f p0 = pack[j + lrow];        // { ||c||^2 , w } for col j+lrow
        const v2f p1 = pack[j + 16 + lrow];   // { ||c||^2 , w } for col j+16+lrow

        // Two independent 16-deep WMMA accumulation chains (K = 64 = 16 x 4)
        v8f c0 = {}; v8f c1 = {};
#pragma unroll
        for (int kk = 0; kk < 16; ++kk) {
            c0 = __builtin_amdgcn_wmma_f32_16x16x4_f32(
                     false, a[kk], false, b0[kk], (short)0, c0, false, false);
            c1 = __builtin_amdgcn_wmma_f32_16x16x4_f32(
                     false, a[kk], false, b1[kk], (short)0, c1, false, false);
        }

#pragma unroll
        for (int r = 0; r < 8; ++r) {
            // C layout: reg r, lane = n + 16*half  ->  (row r + 8*half, col n)
            float sq0 = xnr[r] + p0.x - 2.f * c0[r];
            float sq1 = xnr[r] + p1.x - 2.f * c1[r];
            acc[r] += __expf(-sq0) * p0.y + __expf(-sq1) * p1.y;
        }
    }

    // Reduce over the 16 columns held by each half-wave; write per-split partials.
#pragma unroll
    for (int r = 0; r < 8; ++r) {
        float v = acc[r];
        v += __shfl_xor(v, 1, 32);
        v += __shfl_xor(v, 2, 32);
        v += __shfl_xor(v, 4, 32);
        v += __shfl_xor(v, 8, 32);
        if (lrow == 0)
            partial[split * KROWS + rowBase + r + 8 * half] = v;
    }
}

__global__ __launch_bounds__(256, 1)
void rbf_finish_kernel(const float* __restrict__ partial,
                       const float* __restrict__ bias,
                       float* __restrict__ out)
{
    int i = blockIdx.x * 256 + threadIdx.x;
    float s = bias[0];
#pragma unroll
    for (int p = 0; p < NSPLIT; ++p)
        s += partial[p * KROWS + i];
    out[i] = 1.f / (1.f + __expf(-s));
}

extern "C" void kernel_launch(void* const* d_in, const int* in_sizes, int n_in,
                              void* d_out, int out_size, void* d_ws, size_t ws_size,
                              hipStream_t stream) {
    (void)in_sizes; (void)n_in; (void)out_size; (void)ws_size;
    const float* x  = (const float*)d_in[0];   // [K, 64]
    const float* xb = (const float*)d_in[1];   // [N, 64]
    const float* w  = (const float*)d_in[2];   // [N]
    const float* b  = (const float*)d_in[3];   // [1]
    float* out = (float*)d_out;                // [K]

    float* partial = (float*)d_ws;                          // NSPLIT*KROWS floats (128 KB)
    v2f*   pack    = (v2f*)((char*)d_ws + NSPLIT * KROWS * sizeof(float)); // 64 KB

    rbf_prep_kernel<<<dim3(NCOLS / 256), dim3(256), 0, stream>>>(xb, w, pack);

    const int totalWaves = (KROWS / 16) * NSPLIT;          // 2048 waves
    dim3 grid(totalWaves / WAVES_PER_BLOCK);               // 512 blocks
    dim3 block(WAVES_PER_BLOCK * 32);                      // 128 threads (4 waves)
    rbf_partial_kernel<<<grid, block, 0, stream>>>(x, xb, pack, partial);

    rbf_finish_kernel<<<dim3(KROWS / 256), dim3(256), 0, stream>>>(partial, b, out);
}